// TopKTopPSampler_60043642798747
// MI455X (gfx1250) — compile-verified
//
#include <hip/hip_runtime.h>
#include <stdint.h>

#define BB      256
#define NN      128000
#define TILE    4096
#define SORT_N  4096
#define NTHR    1024
#define NTILES  ((NN + TILE - 1) / TILE)   // 32

typedef float        f4  __attribute__((ext_vector_type(4)));
typedef unsigned int v4u __attribute__((ext_vector_type(4)));
typedef int          v8i __attribute__((ext_vector_type(8)));
typedef int          v4i __attribute__((ext_vector_type(4)));

// Order-preserving float -> uint key (larger float => larger key)
__device__ __forceinline__ unsigned f2key(float x) {
    unsigned u = __float_as_uint(x);
    return u ^ ((unsigned)((int)u >> 31) | 0x80000000u);
}
__device__ __forceinline__ float key2f(unsigned k) {
    unsigned u = (k & 0x80000000u) ? (k ^ 0x80000000u) : ~k;
    return __uint_as_float(u);
}

struct SmemA {
    union {
        float stage[2][TILE];                 // 32 KB TDM staging (phase 1)
        unsigned long long cand[SORT_N];      // 32 KB candidate sort buffer (later)
    } u;
    unsigned hist[4096];                      // 16 KB
    unsigned uaux[NTHR];                      // 4 KB
    float    faux[NTHR];                      // 4 KB
    unsigned cnt, beta, kprime, mcount;
};

// Issue one TDM 1D tile copy: rem elements available from gptr (OOB -> zeros),
// tile_dim0 = TILE fp32 elements, into LDS at lds_off. D# layout per CDNA5 ISA §8.
// This toolchain exposes the 6-arg builtin: (g0 v4u, g1 v8i, g2 v4i, g3 v4i, v8i, cpol).
__device__ __forceinline__ void tdm_issue(const float* gptr, unsigned rem, unsigned lds_off) {
    unsigned long long ga = (unsigned long long)(uintptr_t)gptr;
    v4u g0;
    g0.x = 1u;                                        // count=1, user descriptor
    g0.y = lds_off;                                   // lds_addr (bytes)
    g0.z = (unsigned)(ga & 0xFFFFFFFFu);              // global_addr[31:0]
    g0.w = (unsigned)((ga >> 32) & 0x01FFFFFFu) | (2u << 30); // addr[56:32] | type=2
    v8i g1;
    g1[0] = (int)(2u << 16);                          // data_size = 4B
    g1[1] = (int)((rem & 0xFFFFu) << 16);             // tensor_dim0[15:0]
    g1[2] = (int)((rem >> 16) | (1u << 16));          // tensor_dim0[31:16] | tensor_dim1=1
    g1[3] = (int)((unsigned)TILE << 16);              // tile_dim0
    g1[4] = 1;                                        // tile_dim1=1, tile_dim2=0
    g1[5] = (int)rem;                                 // tensor_dim0_stride (unused for 1 row)
    g1[6] = 0; g1[7] = 0;
    v4i gz4; gz4[0] = 0; gz4[1] = 0; gz4[2] = 0; gz4[3] = 0;
    v8i gz8; gz8[0] = 0; gz8[1] = 0; gz8[2] = 0; gz8[3] = 0;
    gz8[4] = 0; gz8[5] = 0; gz8[6] = 0; gz8[7] = 0;
    __builtin_amdgcn_tensor_load_to_lds(g0, g1, gz4, gz4, gz8, 0);
}

__global__ __launch_bounds__(NTHR, 1) void topk_topp_fused(
    const float* __restrict__ logits, const float* __restrict__ pv,
    const int* __restrict__ kv, float* __restrict__ out) {
    __shared__ SmemA sm;
    const int tid = threadIdx.x;
    const int b   = blockIdx.x;
    const float* row = logits + (size_t)b * NN;

    for (int i = tid; i < 4096; i += NTHR) sm.hist[i] = 0u;
    if (tid == 0) { sm.cnt = 0u; sm.beta = 0u; sm.kprime = 0u; sm.mcount = 0u; }

    int kk = kv[b]; if (kk < 1) kk = 1; if (kk > NN) kk = NN;
    float pp = pv[b];

    // ---- Phase 1: histogram of top-12 key bits, TDM double-buffered staging ----
    if (tid < 32)  // wave 0 drives the Tensor Data Mover
        tdm_issue(row, (unsigned)NN, (unsigned)(uintptr_t)&sm.u.stage[0][0]);
    __syncthreads();
    for (int t = 0; t < NTILES; ++t) {
        int cur = t & 1;
        if (tid < 32) {
            if (t + 1 < NTILES) {
                tdm_issue(row + (size_t)(t + 1) * TILE, (unsigned)(NN - (t + 1) * TILE),
                          (unsigned)(uintptr_t)&sm.u.stage[(t + 1) & 1][0]);
                __builtin_amdgcn_s_wait_tensorcnt(1);  // oldest (tile t) complete
            } else {
                __builtin_amdgcn_s_wait_tensorcnt(0);
            }
        }
        __syncthreads();
        int valid = NN - t * TILE; if (valid > TILE) valid = TILE;
        for (int i = tid; i < valid; i += NTHR) {
            unsigned key = f2key(sm.u.stage[cur][i]);
            atomicAdd(&sm.hist[key >> 20], 1u);
        }
        __syncthreads();
    }

    // ---- Phase 2: block suffix-scan of 4096 bins -> bin beta holding k-th largest ----
    unsigned c0 = sm.hist[4 * tid + 0], c1 = sm.hist[4 * tid + 1];
    unsigned c2 = sm.hist[4 * tid + 2], c3 = sm.hist[4 * tid + 3];
    sm.uaux[tid] = c0 + c1 + c2 + c3;
    __syncthreads();
    for (int off = 1; off < NTHR; off <<= 1) {
        unsigned add = (tid + off < NTHR) ? sm.uaux[tid + off] : 0u;
        __syncthreads();
        sm.uaux[tid] += add;
        __syncthreads();
    }
    {
        unsigned above = (tid < NTHR - 1) ? sm.uaux[tid + 1] : 0u;
        unsigned s3 = c3 + above, s2 = c2 + s3, s1 = c1 + s2, s0 = c0 + s1;
        int best = -1;
        if      (s3 >= (unsigned)kk) best = 4 * tid + 3;
        else if (s2 >= (unsigned)kk) best = 4 * tid + 2;
        else if (s1 >= (unsigned)kk) best = 4 * tid + 1;
        else if (s0 >= (unsigned)kk) best = 4 * tid;
        if (best >= 0) atomicMax(&sm.beta, (unsigned)best);
    }
    __syncthreads();
    unsigned beta = sm.beta;

    // ---- Phase 3: gather (key,index) with bin >= beta (row is L2-hot) ----
    for (int i = tid; i < SORT_N; i += NTHR) sm.u.cand[i] = 0ull;
    __syncthreads();
    const f4* rowv = (const f4*)row;
    for (int i = tid; i < NN / 4; i += NTHR) {
        f4 v = rowv[i];
#pragma unroll
        for (int c = 0; c < 4; ++c) {
            unsigned key = f2key(v[c]);
            if ((key >> 20) >= beta) {
                unsigned pos = atomicAdd(&sm.cnt, 1u);
                if (pos < SORT_N)
                    sm.u.cand[pos] = ((unsigned long long)key << 32) | (unsigned)(4 * i + c);
            }
        }
    }
    __syncthreads();
    unsigned cntc = sm.cnt; if (cntc > SORT_N) cntc = SORT_N;

    // ---- Phase 4: bitonic sort 4096 x 64-bit, descending ----
    for (unsigned size = 2; size <= SORT_N; size <<= 1) {
        for (unsigned strd = size >> 1; strd > 0; strd >>= 1) {
            for (unsigned i = tid; i < SORT_N; i += NTHR) {
                unsigned j = i ^ strd;
                if (j > i) {
                    unsigned long long a = sm.u.cand[i], d = sm.u.cand[j];
                    bool desc = ((i & size) == 0);
                    if (desc ? (a < d) : (a > d)) { sm.u.cand[i] = d; sm.u.cand[j] = a; }
                }
            }
            __syncthreads();
        }
    }

    // ---- Phase 5: exact top-k threshold and tie-extended count k' ----
    unsigned kuse = ((unsigned)kk < cntc) ? (unsigned)kk : cntc;
    if (kuse < 1u) kuse = 1u;
    unsigned Tkey = (unsigned)(sm.u.cand[kuse - 1] >> 32);
    {
        int local = 0;
        for (unsigned i = tid; i < SORT_N; i += NTHR)
            if ((unsigned)(sm.u.cand[i] >> 32) >= Tkey) local++;
        atomicAdd(&sm.kprime, (unsigned)local);
    }
    __syncthreads();
    unsigned kprime = sm.kprime;

    // ---- Phase 6: softmax prefix scan (descending order) -> top-p cutoff ----
    float M = key2f((unsigned)(sm.u.cand[0] >> 32));
    float e[4]; float run = 0.f;
#pragma unroll
    for (int c = 0; c < 4; ++c) {
        unsigned j = 4 * (unsigned)tid + (unsigned)c;
        float v = key2f((unsigned)(sm.u.cand[j] >> 32));
        float ev = (j < kprime) ? __expf(v - M) : 0.f;
        e[c] = run; run += ev;                     // e[c] = local exclusive prefix
    }
    sm.faux[tid] = run;
    __syncthreads();
    for (int off = 1; off < NTHR; off <<= 1) {
        float add = (tid >= off) ? sm.faux[tid - off] : 0.f;
        __syncthreads();
        sm.faux[tid] += add;
        __syncthreads();
    }
    float Z    = sm.faux[NTHR - 1];
    float base = (tid > 0) ? sm.faux[tid - 1] : 0.f;
    float thr  = pp * Z;                           // keep while excl-mass < p * Z
    {
        int local = 0;
#pragma unroll
        for (int c = 0; c < 4; ++c) {
            unsigned j = 4 * (unsigned)tid + (unsigned)c;
            if (j < kprime && (base + e[c]) < thr) local++;
        }
        atomicAdd(&sm.mcount, (unsigned)local);
    }
    __syncthreads();
    unsigned m = sm.mcount;
    if (m < 1u) m = 1u;
    if (kprime > 0u && m > kprime) m = kprime;
    unsigned long long cut = sm.u.cand[m - 1];     // 64-bit (value,index) cutoff

    // ---- Phase 7: streamed filter, non-temporal B128 stores ----
    f4* outv = (f4*)(out + (size_t)b * NN);
    const float ninf = -__builtin_inff();
    for (int i = tid; i < NN / 4; i += NTHR) {
        f4 v = __builtin_nontemporal_load(&rowv[i]);
        f4 o;
#pragma unroll
        for (int c = 0; c < 4; ++c) {
            unsigned long long key64 =
                ((unsigned long long)f2key(v[c]) << 32) | (unsigned)(4 * i + c);
            o[c] = (key64 >= cut) ? v[c] : ninf;
        }
        __builtin_nontemporal_store(o, &outv[i]);
    }
}

extern "C" void kernel_launch(void* const* d_in, const int* in_sizes, int n_in,
                              void* d_out, int out_size, void* d_ws, size_t ws_size,
                              hipStream_t stream) {
    (void)in_sizes; (void)n_in; (void)out_size; (void)d_ws; (void)ws_size;
    const float* logits = (const float*)d_in[0];
    const float* p      = (const float*)d_in[1];
    const int*   k      = (const int*)d_in[2];
    float*       out    = (float*)d_out;
    topk_topp_fused<<<BB, NTHR, 0, stream>>>(logits, p, k, out);
}